// PillarEncoder_4277787427173
// MI455X (gfx1250) — compile-verified
//
#include <hip/hip_runtime.h>
#include <hip/hip_bf16.h>

typedef __attribute__((ext_vector_type(2))) float v2f;
typedef __attribute__((ext_vector_type(8))) float v8f;

#define N_PTS   32
#define K_PAD   12          // K=9 padded to 12 (3 WMMA K-steps of 4)
#define X_L     432
#define Y_L     496
#define C_OUT   64
#define BATCH   4
#define VX      0.16f
#define VY      0.16f
#define X_OFF   0.08f
#define Y_OFF   (0.08f - 39.68f)
#define BN_EPS  1e-3f

// ---------------------------------------------------------------------------
// Kernel 1: zero the BEV canvas (219 MB -> the true bandwidth cost of this op)
// ---------------------------------------------------------------------------
__global__ void pillar_zero_canvas(float4* __restrict__ out, int n4) {
    int i = blockIdx.x * blockDim.x + threadIdx.x;
    int stride = gridDim.x * blockDim.x;
    float4 z = make_float4(0.f, 0.f, 0.f, 0.f);
    for (; i < n4; i += stride) out[i] = z;
}

// ---------------------------------------------------------------------------
// Kernel 2: one wave32 per pillar.
//   features[32x12] built in LDS -> 24x V_WMMA_F32_16X16X4_F32
//   (M-tiles=2 for 32 points, N-tiles=4 for 64 channels, K-steps=3)
//   BN+ReLU+max over points, scatter 64 channels into canvas [B,C,Y,X].
// ---------------------------------------------------------------------------
__global__ __launch_bounds__(32)
void pillar_encode_wmma(const float* __restrict__ pillars,   // [P,32,4]
                        const int*   __restrict__ coors,     // [P,3] (b,x,y)
                        const int*   __restrict__ npoints,   // [P]
                        const float* __restrict__ conv_w,    // [64,9]
                        const float* __restrict__ bn_gamma,
                        const float* __restrict__ bn_beta,
                        const float* __restrict__ bn_mean,
                        const float* __restrict__ bn_var,
                        float*       __restrict__ out)       // [4,64,496,432]
{
    __shared__ float feat[N_PTS][K_PAD];

    const int p    = blockIdx.x;
    const int lane = threadIdx.x;          // 0..31, one point per lane
    const int lm   = lane & 15;
    const bool hi  = lane >= 16;

    // ---- load this lane's point (x,y,z,w) -------------------------------
    const float4 pt4 = ((const float4*)pillars)[(size_t)p * N_PTS + lane];
    const float x = pt4.x, y = pt4.y, z = pt4.z, w = pt4.w;

    // ---- wave32 sum for centroid (reference sums ALL N points) ----------
    float sx = x, sy = y, sz = z;
    #pragma unroll
    for (int off = 16; off >= 1; off >>= 1) {
        sx += __shfl_xor(sx, off, 32);
        sy += __shfl_xor(sy, off, 32);
        sz += __shfl_xor(sz, off, 32);
    }
    const int   np  = npoints[p];
    const float inv = 1.0f / (float)np;
    const float ctx = sx * inv, cty = sy * inv, ctz = sz * inv;

    const int bi = coors[p * 3 + 0];
    const int xi = coors[p * 3 + 1];
    const int yi = coors[p * 3 + 2];
    const float cgx = (float)xi * VX + X_OFF;
    const float cgy = (float)yi * VY + Y_OFF;

    // ---- 9 features, masked, K padded to 12 with zeros ------------------
    const float m = (lane < np) ? 1.0f : 0.0f;
    feat[lane][0]  = x * m;
    feat[lane][1]  = y * m;
    feat[lane][2]  = z * m;
    feat[lane][3]  = w * m;
    feat[lane][4]  = (x - ctx) * m;
    feat[lane][5]  = (y - cty) * m;
    feat[lane][6]  = (z - ctz) * m;
    feat[lane][7]  = (x - cgx) * m;
    feat[lane][8]  = (y - cgy) * m;
    feat[lane][9]  = 0.f;
    feat[lane][10] = 0.f;
    feat[lane][11] = 0.f;
    __syncthreads();

    // ---- A fragments: 16x4 f32, lane L: M=L%16, K = {0,1}/{2,3} + 4*step
    v2f afrag[2][3];
    #pragma unroll
    for (int ks = 0; ks < 3; ++ks) {
        const int k0 = 4 * ks + (hi ? 2 : 0);
        #pragma unroll
        for (int mt = 0; mt < 2; ++mt) {
            const int row = 16 * mt + lm;
            afrag[mt][ks].x = feat[row][k0];
            afrag[mt][ks].y = feat[row][k0 + 1];
        }
    }

    // ---- B fragments: 4x16 f32, lane L: N=L%16 (+16*tile), same K split
    v2f bfrag[4][3];
    #pragma unroll
    for (int nt = 0; nt < 4; ++nt) {
        const int ch = 16 * nt + lm;
        #pragma unroll
        for (int ks = 0; ks < 3; ++ks) {
            const int k0 = 4 * ks + (hi ? 2 : 0);
            bfrag[nt][ks].x = (k0     < 9) ? conv_w[ch * 9 + k0]     : 0.f;
            bfrag[nt][ks].y = (k0 + 1 < 9) ? conv_w[ch * 9 + k0 + 1] : 0.f;
        }
    }

    // ---- 24x v_wmma_f32_16x16x4_f32 -------------------------------------
    v8f acc[2][4];
    #pragma unroll
    for (int mt = 0; mt < 2; ++mt) {
        #pragma unroll
        for (int nt = 0; nt < 4; ++nt) {
            v8f d = {0.f, 0.f, 0.f, 0.f, 0.f, 0.f, 0.f, 0.f};
            #pragma unroll
            for (int ks = 0; ks < 3; ++ks) {
                d = __builtin_amdgcn_wmma_f32_16x16x4_f32(
                        /*neg_a=*/false, afrag[mt][ks],
                        /*neg_b=*/false, bfrag[nt][ks],
                        /*c_mod=*/(short)0, d,
                        /*reuse_a=*/false, /*reuse_b=*/false);
            }
            acc[mt][nt] = d;
        }
    }

    // ---- BN + ReLU + max over points (M dim) ----------------------------
    // D layout: lane L holds column N=L%16, rows M = vgpr + (L>=16 ? 8 : 0).
    // ReLU output >= 0, so 0 is a valid identity for the max.
    float pooled[4];
    #pragma unroll
    for (int nt = 0; nt < 4; ++nt) {
        const int ch = 16 * nt + lm;
        const float scale = bn_gamma[ch] * rsqrtf(bn_var[ch] + BN_EPS);
        const float shift = bn_beta[ch] - bn_mean[ch] * scale;
        float mx = 0.0f;
        #pragma unroll
        for (int mt = 0; mt < 2; ++mt) {
            #pragma unroll
            for (int i = 0; i < 8; ++i) {
                float v = acc[mt][nt][i] * scale + shift;
                mx = fmaxf(mx, fmaxf(v, 0.0f));
            }
        }
        mx = fmaxf(mx, __shfl_xor(mx, 16, 32));   // merge M row-halves
        pooled[nt] = mx;
    }

    // ---- scatter: out[b, ch, y, x] --------------------------------------
    if (lane < 16) {
        const size_t base = (size_t)bi * C_OUT * Y_L * X_L;
        #pragma unroll
        for (int nt = 0; nt < 4; ++nt) {
            const int ch = 16 * nt + lm;
            out[base + ((size_t)ch * Y_L + yi) * X_L + xi] = pooled[nt];
        }
    }
}

extern "C" void kernel_launch(void* const* d_in, const int* in_sizes, int n_in,
                              void* d_out, int out_size, void* d_ws, size_t ws_size,
                              hipStream_t stream) {
    const float* pillars  = (const float*)d_in[0];
    const int*   coors    = (const int*)  d_in[1];
    const int*   npoints  = (const int*)  d_in[2];
    const float* conv_w   = (const float*)d_in[3];
    const float* bn_gamma = (const float*)d_in[4];
    const float* bn_beta  = (const float*)d_in[5];
    const float* bn_mean  = (const float*)d_in[6];
    const float* bn_var   = (const float*)d_in[7];
    float* out = (float*)d_out;

    const int P = in_sizes[2];             // 40000 pillars

    // 1) zero the canvas (bandwidth-dominant phase): float4 grid-stride
    const int n4 = out_size / 4;
    pillar_zero_canvas<<<2048, 256, 0, stream>>>((float4*)out, n4);

    // 2) one wave32 per pillar: WMMA encode + pool + scatter
    pillar_encode_wmma<<<P, 32, 0, stream>>>(
        pillars, coors, npoints, conv_w,
        bn_gamma, bn_beta, bn_mean, bn_var, out);
}